// gt_edge_19894288515366
// MI455X (gfx1250) — compile-verified
//
#include <hip/hip_runtime.h>
#include <stdint.h>

// out = (sum_k |sobel_k * x| > 0) ? 1 : 0   (sigmoid>0.5 <=> s>0, s = sum of abs >= 0)
// HBM-bound: 201 MB @ 23.3 TB/s ~ 8.6 us floor. Strategy: TDM (tensor_load_to_lds)
// stages one halo tile per block into LDS, factored VALU sobel, NT vector stores.

typedef __attribute__((ext_vector_type(4))) float    f4;
typedef __attribute__((ext_vector_type(2))) float    f2;
typedef __attribute__((ext_vector_type(4))) unsigned u32x4;
typedef __attribute__((ext_vector_type(8))) unsigned u32x8;

#define IMG_W 512
#define IMG_H 512
#define TW 64            // tile output width
#define TH 16            // tile output height
#define HALO_W (TW + 2)  // 66
#define HALO_H (TH + 2)  // 18
#define PITCH 68         // LDS row pitch in floats: 68*4 = 272 = 17*16 -> rows 16B aligned

__global__ __launch_bounds__(256)
void gt_edge_sobel_kernel(const float* __restrict__ src, float* __restrict__ dst)
{
    __shared__ __align__(16) float tile[HALO_H * PITCH];

    const int t  = threadIdx.x;
    const int x0 = blockIdx.x * TW;
    const int y0 = blockIdx.y * TH;
    const size_t img_off = (size_t)blockIdx.z * (size_t)(IMG_W * IMG_H);
    const float* s = src + img_off;
    float*       d = dst + img_off;

    // ---- zero-fill LDS: cells TDM doesn't touch (image borders) => SAME zero padding ----
    {
        f4 z = (f4)0.0f;
        f4* tp = (f4*)tile;
        #pragma unroll
        for (int i = t; i < (HALO_H * PITCH) / 4; i += 256)
            tp[i] = z;
    }
    __syncthreads();

    // ---- TDM: one tensor_load_to_lds per block (wave 0 only; TDM is per-wave) ----
    if (t < 32) {
        // Clip halo rect at image borders; right-edge OOB is zero-filled by tensor_dim0,
        // bottom edge handled by shortening the iteration count.
        const int clip_left = (x0 == 0) ? 1 : 0;
        const int clip_top  = (y0 == 0) ? 1 : 0;
        const int clip_bot  = (y0 + TH == IMG_H) ? 1 : 0;
        const int start_x   = x0 - 1 + clip_left;
        const int start_y   = y0 - 1 + clip_top;
        const unsigned load_w = HALO_W - clip_left;              // elements per row (tile_dim0)
        const unsigned load_h = HALO_H - clip_top - clip_bot;    // rows (iterations)
        const unsigned td0    = (unsigned)(IMG_W - start_x);     // right-edge OOB -> zeros

        const uint64_t ga = (uint64_t)(uintptr_t)(s + (size_t)start_y * IMG_W + start_x);
        // low 32 bits of a generic pointer to __shared__ == LDS byte offset
        const unsigned lds_base =
            (unsigned)(uintptr_t)(&tile[clip_top * PITCH + clip_left]);

        // ---- D# group 0: count=1 | lds_addr | global_addr[56:0] | type=2 ----
        u32x4 g0;
        g0.x = 1u;                                           // count=1, user mode
        g0.y = lds_base;                                     // lds_addr (bytes)
        g0.z = (unsigned)ga;                                 // global_addr[31:0]
        g0.w = ((unsigned)(ga >> 32) & 0x01FFFFFFu)          // global_addr[56:32]
             | (2u << 30);                                   // type=2 ("image")
        // ---- D# group 1: data_size=4B, iterate, dims/strides ----
        u32x8 g1;
        g1.s0 = (2u << 16)                                   // data_size = 4 bytes
              | (1u << 19);                                  // iterate_enable
        g1.s1 = (td0 & 0xFFFFu) << 16;                       // tensor_dim0[15:0]
        g1.s2 = (td0 >> 16) | (1u << 16);                    // tensor_dim0[31:16] | tensor_dim1=1
        g1.s3 = load_w << 16;                                // tile_dim0
        g1.s4 = 1u;                                          // tile_dim1=1 (one row per iter)
        g1.s5 = (unsigned)IMG_W;                             // tensor_dim0_stride
        g1.s6 = 0u;
        g1.s7 = 0u;
        // ---- D# group 2 (iterate mode): lds+=68 dw, global+=512 dw per row ----
        u32x4 g2;
        g2.x = 0u;                                           // tensor_dim2 (unused)
        g2.y = (unsigned)PITCH;                              // lds_addr_increment (data_size units)
        g2.z = (unsigned)IMG_W;                              // global_addr_increment (data_size units)
        g2.w = (load_h - 1u) << 16;                          // iterate_count (0 => 1x)
        // ---- D# group 3: unused higher dims ----
        u32x4 g3;
        g3.x = 0u; g3.y = 0u; g3.z = 0u; g3.w = 0u;

        asm volatile("tensor_load_to_lds %0, %1, %2, %3"
                     :: "s"(g0), "s"(g1), "s"(g2), "s"(g3)
                     : "memory");
#if defined(__has_builtin) && __has_builtin(__builtin_amdgcn_s_wait_tensorcnt)
        __builtin_amdgcn_s_wait_tensorcnt(0);
#else
        asm volatile("s_wait_tensorcnt 0x0" ::: "memory");
#endif
    }
    __syncthreads();

    // ---- compute: each thread does 1 row x 4 consecutive pixels ----
    const int r  = t >> 4;          // 0..15  (output row in tile)
    const int c4 = (t & 15) << 2;   // 0,4,...,60 (first output col in tile)

    const float* row0 = &tile[(r + 0) * PITCH + c4];
    const float* row1 = &tile[(r + 1) * PITCH + c4];
    const float* row2 = &tile[(r + 2) * PITCH + c4];

    // 16B-aligned b128 + 8B-aligned b64 LDS reads (6 taps per row)
    f4 a0v = *(const f4*)row0;        f2 a0w = *(const f2*)(row0 + 4);
    f4 a1v = *(const f4*)row1;        f2 a1w = *(const f2*)(row1 + 4);
    f4 a2v = *(const f4*)row2;        f2 a2w = *(const f2*)(row2 + 4);

    float a0[6] = {a0v.x, a0v.y, a0v.z, a0v.w, a0w.x, a0w.y};
    float a1[6] = {a1v.x, a1v.y, a1v.z, a1v.w, a1w.x, a1w.y};
    float a2[6] = {a2v.x, a2v.y, a2v.z, a2v.w, a2w.x, a2w.y};

    // column-hoisted terms shared by the 4 output pixels
    float colA[6], colB[6];
    #pragma unroll
    for (int c = 0; c < 6; ++c) {
        colB[c] = a0[c] - a2[c];                  // vertical diff   (k1)
        colA[c] = a0[c] + 2.0f * a1[c] + a2[c];   // vertical smooth (k0)
    }

    float res[4];
    #pragma unroll
    for (int j = 0; j < 4; ++j) {
        float u  = a1[j] - a1[j + 2];                              // mid-row diff (k2,k3)
        float e0 = colA[j + 2] - colA[j];                          // [-1 0 1;-2 0 2;-1 0 1]
        float e1 = colB[j] + 2.0f * colB[j + 1] + colB[j + 2];     // [1 2 1; 0 0 0;-1 -2 -1]
        float e2 = 2.0f * (a0[j] - a2[j + 2]) + (colB[j + 1] + u); // [2 1 0; 1 0 -1; 0 -1 -2]
        float e3 = 2.0f * (a2[j] - a0[j + 2]) + (u - colB[j + 1]); // [0 -1 -2; 1 0 -1; 2 1 0]
        float sum = __builtin_fabsf(e0) + __builtin_fabsf(e1)
                  + __builtin_fabsf(e2) + __builtin_fabsf(e3);
        res[j] = (sum > 0.0f) ? 1.0f : 0.0f;
    }

    f4 o;
    o.x = res[0]; o.y = res[1]; o.z = res[2]; o.w = res[3];

    // non-temporal 16B store: keep 100MB output from thrashing L2 (input fits in 192MB L2)
    float* outp = d + (size_t)(y0 + r) * IMG_W + (x0 + c4);
    __builtin_nontemporal_store(o, (f4*)outp);
}

extern "C" void kernel_launch(void* const* d_in, const int* in_sizes, int n_in,
                              void* d_out, int out_size, void* d_ws, size_t ws_size,
                              hipStream_t stream)
{
    const float* x = (const float*)d_in[0];
    float* out = (float*)d_out;

    const int n_images = in_sizes[0] / (IMG_W * IMG_H);  // B*C = 96
    dim3 grid(IMG_W / TW, IMG_H / TH, n_images);         // (8, 32, 96)
    gt_edge_sobel_kernel<<<grid, 256, 0, stream>>>(x, out);
}